// VAELoss_27273042329742
// MI455X (gfx1250) — compile-verified
//
#include <hip/hip_runtime.h>
#include <hip/hip_bf16.h>

// Problem dims (from reference)
#define B_DIM 1024
#define S_DIM 277
#define R_DIM 76
#define Z_DIM 56
#define NROWS (B_DIM * S_DIM)

typedef __attribute__((ext_vector_type(2))) float v2f;
typedef __attribute__((ext_vector_type(8))) float v8f;

// ---------------- wave32 reductions (CDNA5 is wave32-only) ----------------
__device__ __forceinline__ float wave_fmax(float v) {
#pragma unroll
    for (int o = 16; o > 0; o >>= 1) v = fmaxf(v, __shfl_xor(v, o, 32));
    return v;
}
__device__ __forceinline__ float wave_fmin(float v) {
#pragma unroll
    for (int o = 16; o > 0; o >>= 1) v = fminf(v, __shfl_xor(v, o, 32));
    return v;
}
__device__ __forceinline__ float wave_fsum(float v) {
#pragma unroll
    for (int o = 16; o > 0; o >>= 1) v += __shfl_xor(v, o, 32);
    return v;
}
__device__ __forceinline__ int wave_imax(int v) {
#pragma unroll
    for (int o = 16; o > 0; o >>= 1) {
        int t = __shfl_xor(v, o, 32);
        v = (t > v) ? t : v;
    }
    return v;
}

// ---------------- pass 1: global min/max of model_out_x -------------------
__global__ void __launch_bounds__(256)
minmax_partial(const float* __restrict__ x, long n4,
               float* __restrict__ pmin, float* __restrict__ pmax) {
    const float4* x4 = (const float4*)x;
    float mn = INFINITY, mx = -INFINITY;
    for (long i = (long)blockIdx.x * blockDim.x + threadIdx.x; i < n4;
         i += (long)gridDim.x * blockDim.x) {
        float4 v = x4[i];
        mn = fminf(mn, fminf(fminf(v.x, v.y), fminf(v.z, v.w)));
        mx = fmaxf(mx, fmaxf(fmaxf(v.x, v.y), fmaxf(v.z, v.w)));
    }
    mn = wave_fmin(mn);
    mx = wave_fmax(mx);
    __shared__ float smn[8], smx[8];
    const int wid = threadIdx.x >> 5, lane = threadIdx.x & 31;
    if (lane == 0) { smn[wid] = mn; smx[wid] = mx; }
    __syncthreads();
    if (threadIdx.x == 0) {
        float a = smn[0], b = smx[0];
        for (int w = 1; w < 8; ++w) { a = fminf(a, smn[w]); b = fmaxf(b, smx[w]); }
        pmin[blockIdx.x] = a;
        pmax[blockIdx.x] = b;
    }
}

__global__ void __launch_bounds__(256)
minmax_final(const float* __restrict__ pmin, const float* __restrict__ pmax,
             int nb, float* __restrict__ shift_out) {
    float mn = INFINITY, mx = -INFINITY;
    for (int i = threadIdx.x; i < nb; i += blockDim.x) {
        mn = fminf(mn, pmin[i]);
        mx = fmaxf(mx, pmax[i]);
    }
    mn = wave_fmin(mn);
    mx = wave_fmax(mx);
    __shared__ float smn[8], smx[8];
    const int wid = threadIdx.x >> 5, lane = threadIdx.x & 31;
    if (lane == 0) { smn[wid] = mn; smx[wid] = mx; }
    __syncthreads();
    if (threadIdx.x == 0) {
        float a = smn[0], b = smx[0];
        for (int w = 1; w < 8; ++w) { a = fminf(a, smn[w]); b = fmaxf(b, smx[w]); }
        shift_out[0] = -100.0f + (a - b);   // -100 + (min - max)
    }
}

// ---------------- pass 2: masked-softmax BCE, one wave32 per row ----------
// lanes cover r = lane, lane+32, lane+64 (third predicated: 64+12 = 76)
__global__ void __launch_bounds__(256)
bce_kernel(const float* __restrict__ model, const float* __restrict__ target,
           const float* __restrict__ masks, const int* __restrict__ ind2lhs,
           const float* __restrict__ shift_p, float* __restrict__ partials) {
    const float shift = shift_p[0];
    const int lane = threadIdx.x & 31;
    const int wid  = threadIdx.x >> 5;
    const int gwave = blockIdx.x * (blockDim.x >> 5) + wid;
    const int nwave = gridDim.x * (blockDim.x >> 5);
    const int r2 = lane + 64;
    const bool v2 = (r2 < R_DIM);

    float acc = 0.0f;
    for (int row = gwave; row < NROWS; row += nwave) {
        const size_t base = (size_t)row * R_DIM;
        const float t0 = target[base + lane];
        const float t1 = target[base + lane + 32];
        const float t2 = v2 ? target[base + r2] : 0.0f;
        float x0 = model[base + lane];
        float x1 = model[base + lane + 32];
        float x2 = v2 ? model[base + r2] : 0.0f;

        // one-hot argmax -> true rule index (wave max over candidate idx)
        int idx = -1;
        if (t0 > 0.5f) idx = lane;
        if (t1 > 0.5f) idx = lane + 32;
        if (v2 && t2 > 0.5f) idx = r2;
        idx = wave_imax(idx);

        const int lhs = ind2lhs[idx];
        const float* mrow = masks + (size_t)lhs * R_DIM;
        x0 += (1.0f - mrow[lane]) * shift;
        x1 += (1.0f - mrow[lane + 32]) * shift;
        if (v2) x2 += (1.0f - mrow[r2]) * shift;

        float mx = fmaxf(x0, x1);
        if (v2) mx = fmaxf(mx, x2);
        mx = wave_fmax(mx);

        const float e0 = expf(x0 - mx);
        const float e1 = expf(x1 - mx);
        const float e2 = v2 ? expf(x2 - mx) : 0.0f;
        const float se = wave_fsum(e0 + e1 + e2);
        const float ls = logf(se);
        const float inv = 1.0f / se;

        float s;
        {
            float lp = fmaxf(x0 - mx - ls, -100.0f);
            float l1 = fmaxf(log1pf(-fminf(e0 * inv, 1.0f)), -100.0f);
            s = t0 * lp + (1.0f - t0) * l1;
        }
        {
            float lp = fmaxf(x1 - mx - ls, -100.0f);
            float l1 = fmaxf(log1pf(-fminf(e1 * inv, 1.0f)), -100.0f);
            s += t1 * lp + (1.0f - t1) * l1;
        }
        if (v2) {
            float lp = fmaxf(x2 - mx - ls, -100.0f);
            float l1 = fmaxf(log1pf(-fminf(e2 * inv, 1.0f)), -100.0f);
            s += t2 * lp + (1.0f - t2) * l1;
        }
        acc += s;   // lane-local; single wave reduction at the end
    }

    acc = wave_fsum(acc);
    __shared__ float sred[8];
    if (lane == 0) sred[wid] = acc;
    __syncthreads();
    if (threadIdx.x == 0) {
        float t = 0.0f;
        for (int w = 0; w < 8; ++w) t += sred[w];
        partials[blockIdx.x] = t;   // fixed-order, deterministic
    }
}

// ---------------- pass 3: mu^T mu via V_WMMA_F32_16X16X4_F32 --------------
// One block, 16 waves; wave w owns 16x16 tile (ti=w>>2, tj=w&3) of the 64x64
// (zero-padded from 56) Gram matrix. mu staged through LDS in 128-row chunks.
// A-layout (ISA 7.12.2, 32-bit 16x4): lanes 0-15 M=lane K={0,1},
// lanes 16-31 M=lane-16 K={2,3}. B mirrors with N in place of M.
__global__ void __launch_bounds__(512)
mom_kernel(const float* __restrict__ mu, float* __restrict__ mom_out) {
    __shared__ float chunk[128 * Z_DIM];   // 28 KB
    __shared__ float red[16];
    __shared__ float cs[64];

    const int tid  = threadIdx.x;
    const int lane = tid & 31;
    const int wid  = tid >> 5;          // 0..15
    const int ti   = wid >> 2;          // tile row 0..3
    const int tj   = wid & 3;           // tile col 0..3
    const int mrow = lane & 15;
    const int kb   = (lane >> 4) << 1;  // 0 or 2
    const int ca   = ti * 16 + mrow;    // A column of mu (M index)
    const int cb   = tj * 16 + mrow;    // B column of mu (N index)
    const bool va  = (ca < Z_DIM);
    const bool vb  = (cb < Z_DIM);

    v8f acc = {0.f, 0.f, 0.f, 0.f, 0.f, 0.f, 0.f, 0.f};
    float colsum = 0.0f;

    for (int c = 0; c < 8; ++c) {
        const int kc = c * 128;
        for (int i = tid; i < 128 * Z_DIM; i += 512)
            chunk[i] = mu[(size_t)kc * Z_DIM + i];
        __syncthreads();

        if (tid < Z_DIM) {                      // avg_mu column sums
            float s = 0.0f;
            for (int rr = 0; rr < 128; ++rr) s += chunk[rr * Z_DIM + tid];
            colsum += s;
        }

        for (int kk = 0; kk < 128; kk += 4) {
            const float* r0 = &chunk[(kk + kb) * Z_DIM];
            const float* r1 = &chunk[(kk + kb + 1) * Z_DIM];
            v2f a, b;
            a.x = va ? r0[ca] : 0.0f;
            a.y = va ? r1[ca] : 0.0f;
            b.x = vb ? r0[cb] : 0.0f;
            b.y = vb ? r1[cb] : 0.0f;
            acc = __builtin_amdgcn_wmma_f32_16x16x4_f32(
                false, a, false, b, (short)0, acc, false, false);
        }
        __syncthreads();
    }

    // D layout: VGPR v -> M = v + 8*(lane>>4), N = lane&15 (plus tile offset)
    float local = 0.0f;
    const int ibase = ti * 16 + ((lane >> 4) << 3);
    const int j = tj * 16 + (lane & 15);
#pragma unroll
    for (int v = 0; v < 8; ++v) {
        const int i = ibase + v;
        if (i < Z_DIM && j < Z_DIM) {
            float var = acc[v] * (1.0f / B_DIM);
            float e = var - ((i == j) ? 1.0f : 0.0f);
            local += tanhf(e) * e;
        }
    }
    local = wave_fsum(local);
    if (lane == 0) red[wid] = local;
    if (tid < 64) cs[tid] = (tid < Z_DIM) ? colsum : 0.0f;
    __syncthreads();
    if (tid == 0) {
        float tot = 0.0f;
        for (int w = 0; w < 16; ++w) tot += red[w];
        float ss = 0.0f;
        for (int t = 0; t < Z_DIM; ++t) {
            float a = cs[t] * (1.0f / B_DIM);
            ss += a * a;
        }
        mom_out[0] = ss * (1.0f / Z_DIM) + tot * (1.0f / (Z_DIM * Z_DIM));
    }
}

// ---------------- pass 4: combine --------------------------------------
__global__ void __launch_bounds__(256)
final_combine(const float* __restrict__ pbce, int nb,
              const float* __restrict__ mom, float* __restrict__ out) {
    float s = 0.0f;
    for (int i = threadIdx.x; i < nb; i += blockDim.x) s += pbce[i];
    s = wave_fsum(s);
    __shared__ float sred[8];
    const int wid = threadIdx.x >> 5, lane = threadIdx.x & 31;
    if (lane == 0) sred[wid] = s;
    __syncthreads();
    if (threadIdx.x == 0) {
        float t = 0.0f;
        for (int w = 0; w < 8; ++w) t += sred[w];
        // BCE = S * (-sum / (B*S*R)) = -sum / (B*R)
        float BCE = -t / ((float)B_DIM * (float)R_DIM);
        out[0] = BCE + mom[0];
    }
}

extern "C" void kernel_launch(void* const* d_in, const int* in_sizes, int n_in,
                              void* d_out, int out_size, void* d_ws, size_t ws_size,
                              hipStream_t stream) {
    (void)in_sizes; (void)n_in; (void)out_size; (void)ws_size;
    const float* model  = (const float*)d_in[0];  // [B,S,R]
    const float* mu     = (const float*)d_in[1];  // [B,Z]
    // d_in[2] = log_var: unused by the loss (sample_z=False, training mode)
    const float* target = (const float*)d_in[3];  // [B,S,R] one-hot
    const float* masks  = (const float*)d_in[4];  // [NLHS,R]
    const int*   i2l    = (const int*)d_in[5];    // [R]
    float* out = (float*)d_out;
    float* ws  = (float*)d_ws;

    float* shift = ws;             // [1]
    float* mom   = ws + 1;         // [1]
    float* pmin  = ws + 16;        // [1024]
    float* pmax  = ws + 16 + 1024; // [1024]
    float* pbce  = ws + 16 + 2048; // [1024]

    const long n4 = (long)B_DIM * S_DIM * R_DIM / 4;   // divisible by 4

    minmax_partial<<<1024, 256, 0, stream>>>(model, n4, pmin, pmax);
    minmax_final<<<1, 256, 0, stream>>>(pmin, pmax, 1024, shift);
    bce_kernel<<<1024, 256, 0, stream>>>(model, target, masks, i2l, shift, pbce);
    mom_kernel<<<1, 512, 0, stream>>>(mu, mom);
    final_combine<<<1, 256, 0, stream>>>(pbce, 1024, mom, out);
}